// Attention_loss_76794015252989
// MI455X (gfx1250) — compile-verified
//
#include <hip/hip_runtime.h>
#include <stdint.h>

// Problem constants (fixed by the reference file)
#define B_IMG     32
#define N_BOX     128
#define AH        512
#define AW        512
#define TILE_ROWS 16
#define N_TILES   (AH / TILE_ROWS)        // 32 tiles per image
#define ROW_PAD   4                        // floats of pad per LDS row (bank stagger)
#define ROW_STR   (AW + ROW_PAD)           // 516 floats per LDS row

// ---------------------------------------------------------------------------
// CDNA5 async global->LDS copy (ASYNCcnt path). The gfx1250 builtin takes
// (AS1 v4i* src, AS3 v4i* dst, imm offset, imm cpol); the compiler diagnostic
// showed param0 as "v4i __device__ *" (HIP prints AS1 as __device__).
// Address-space pointers are produced via integer round-trips: on amdgcn a
// generic LDS pointer's low 32 bits are the LDS byte offset, and
// generic->global is value-identity.
// ---------------------------------------------------------------------------
typedef __attribute__((ext_vector_type(4))) int v4i;
typedef __attribute__((address_space(1))) v4i as1_v4i;
typedef __attribute__((address_space(3))) v4i as3_v4i;

__device__ __forceinline__ void async_copy_b128(const void* gsrc, uint32_t lds_off) {
#if __has_builtin(__builtin_amdgcn_global_load_async_to_lds_b128)
  __builtin_amdgcn_global_load_async_to_lds_b128(
      (as1_v4i*)(uintptr_t)gsrc, (as3_v4i*)(uintptr_t)lds_off,
      /*offset=*/0, /*cpol=*/0);
#else
  asm volatile("global_load_async_to_lds_b128 %0, %1, off"
               :: "v"(lds_off), "v"(gsrc) : "memory");
#endif
}

__device__ __forceinline__ void wait_asynccnt0() {
#if __has_builtin(__builtin_amdgcn_s_wait_asynccnt)
  __builtin_amdgcn_s_wait_asynccnt(0);
#else
  asm volatile("s_wait_asynccnt 0" ::: "memory");
#endif
}

// ---------------------------------------------------------------------------
// Kernel 1: one block per (tile, image). Async-stream the 512x16 f32 tile of
// the attention map into LDS while rasterizing the 128 boxes (VALU-only),
// then BCE out of LDS and reduce to one f32 slot per block (deterministic).
// ---------------------------------------------------------------------------
__global__ __launch_bounds__(256)
void attn_loss_main(const float* __restrict__ att,
                    const float* __restrict__ bboxs,
                    const int*   __restrict__ imgh_p,
                    const int*   __restrict__ imgw_p,
                    float*       __restrict__ partials) {
  __shared__ __align__(16) float pTile[TILE_ROWS * ROW_STR];   // ~33 KB
  __shared__ int   bx1[N_BOX], by1[N_BOX], bx2[N_BOX], by2[N_BOX];
  __shared__ float wx1[N_BOX], wy1[N_BOX], wx2[N_BOX], wy2[N_BOX];
  __shared__ float waveSums[8];

  const int tid   = threadIdx.x;
  const int tileY = blockIdx.x;          // 0..31
  const int img   = blockIdx.y;          // 0..31
  const int row0  = tileY * TILE_ROWS;

  const float imw = (float)imgw_p[0];
  const float imh = (float)imgh_p[0];
  const float sxf = (float)AW / imw;     // 0.25 for the reference sizes
  const float syf = (float)AH / imh;

  // ---- 1) kick off the async DMA of the tile (2048 x 16B chunks total,
  //         8 per thread, consecutive lanes -> consecutive chunks) ----------
  const float* attBase = att + ((size_t)img * AH + (size_t)row0) * AW;
#pragma unroll
  for (int i = 0; i < 8; ++i) {
    const int c  = tid + i * 256;        // chunk index; 128 chunks per row
    const int r  = c >> 7;
    const int jj = c & 127;
    const uint32_t loff = (uint32_t)(uintptr_t)&pTile[r * ROW_STR + jj * 4];
    async_copy_b128(attBase + (size_t)r * AW + jj * 4, loff);
  }

  // ---- 2) preprocess boxes while the DMA is in flight ---------------------
  if (tid < N_BOX) {
    const float* bb = bboxs + ((size_t)img * N_BOX + tid) * 5;
    const float cx1 = bb[0], cy1 = bb[1], cx2 = bb[2], cy2 = bb[3], lab = bb[4];
    const bool valid = (lab != -1.0f) && (cx1 <= imw) && (cy1 <= imh) &&
                       (cx2 <= imw) && (cy2 <= imh);
    const float sx1 = cx1 * sxf, sy1 = cy1 * syf;
    const float sx2 = cx2 * sxf, sy2 = cy2 * syf;
    const float fx1 = floorf(sx1), fy1 = floorf(sy1);
    wx1[tid] = fx1 + 1.0f - sx1;                 // x1m
    wy1[tid] = fy1 + 1.0f - sy1;                 // y1m
    wx2[tid] = sx2 - floorf(sx2);                // x2m
    wy2[tid] = sy2 - floorf(sy2);                // y2m
    bx1[tid] = (int)fmaxf(fx1, 0.0f);
    bx2[tid] = (int)fminf(ceilf(sx2) + 1.0f, (float)AW);
    by1[tid] = valid ? (int)fmaxf(fy1, 0.0f) : 0x7fffffff;  // fold validity
    by2[tid] = (int)fminf(ceilf(sy2) + 1.0f, (float)AH);
  }
  __syncthreads();

  // ---- 3) rasterize boxes (sequential overwrite == scan semantics) -------
  // thread -> row r = tid/16 within tile; columns k, k+16, ..., k+496
  const int r = tid >> 4;
  const int k = tid & 15;
  const int y = row0 + r;

  float vals[32];
#pragma unroll
  for (int i = 0; i < 32; ++i) vals[i] = 0.0f;

  for (int b = 0; b < N_BOX; ++b) {
    const int y1 = by1[b];
    const int y2 = by2[b];
    if (y < y1 || y >= y2) continue;     // also rejects invalid boxes
    const float roww = (y == y1 ? wy1[b] : 1.0f) * (y == y2 - 1 ? wy2[b] : 1.0f);
    const int   x1 = bx1[b], x2 = bx2[b];
    const float x1m = wx1[b], x2m = wx2[b];
#pragma unroll
    for (int i = 0; i < 32; ++i) {
      const int x = k + 16 * i;
      if (x >= x1 && x < x2) {
        const float colw = (x == x1 ? x1m : 1.0f) * (x == x2 - 1 ? x2m : 1.0f);
        vals[i] = roww * colw;           // later boxes overwrite earlier ones
      }
    }
  }

  // ---- 4) wait for the tile, then the BCE pass out of LDS -----------------
  wait_asynccnt0();
  __syncthreads();

  float s = 0.0f;
#pragma unroll
  for (int i = 0; i < 32; ++i) {
    const float p  = pTile[r * ROW_STR + k + 16 * i];
    const float lp = fmaxf(logf(p), -100.0f);
    const float l1 = fmaxf(log1pf(-p), -100.0f);
    const float m  = vals[i];
    s += m * lp + (1.0f - m) * l1;
  }

  // ---- 5) deterministic block reduction -> one slot per (img, tile) -------
  for (int o = 16; o > 0; o >>= 1) s += __shfl_down(s, o, 32);
  if ((tid & 31) == 0) waveSums[tid >> 5] = s;
  __syncthreads();
  if (tid == 0) {
    float t = 0.0f;
#pragma unroll
    for (int w = 0; w < 8; ++w) t += waveSums[w];
    partials[img * N_TILES + tileY] = t;
  }
}

// ---------------------------------------------------------------------------
// Kernel 2: 1024 threads = 32 wave32s; wave w reduces the 32 tile partials of
// image w, gates by any(valid), then thread 0 averages over the batch.
// ---------------------------------------------------------------------------
__global__ __launch_bounds__(1024)
void attn_loss_final(const float* __restrict__ bboxs,
                     const int*   __restrict__ imgh_p,
                     const int*   __restrict__ imgw_p,
                     const float* __restrict__ partials,
                     float*       __restrict__ out) {
  __shared__ float lossArr[B_IMG];
  const int tid = threadIdx.x;
  const int img = tid >> 5;
  const int j   = tid & 31;

  float v = partials[img * N_TILES + j];
  for (int o = 16; o > 0; o >>= 1) v += __shfl_down(v, o, 32);

  if (j == 0) {
    const float imw = (float)imgw_p[0];
    const float imh = (float)imgh_p[0];
    bool any = false;
    const float* bb = bboxs + (size_t)img * N_BOX * 5;
    for (int b = 0; b < N_BOX; ++b) {
      const float* q = bb + b * 5;
      any = any || ((q[4] != -1.0f) && (q[0] <= imw) && (q[1] <= imh) &&
                    (q[2] <= imw) && (q[3] <= imh));
    }
    lossArr[img] = any ? (-v * (1.0f / ((float)AH * (float)AW))) : 0.0f;
  }
  __syncthreads();

  if (tid == 0) {
    float t = 0.0f;
#pragma unroll
    for (int i = 0; i < B_IMG; ++i) t += lossArr[i];
    out[0] = t * (1.0f / (float)B_IMG);
  }
}

// ---------------------------------------------------------------------------
// Host-side launcher (graph-capture safe: only kernel launches on `stream`).
// Inputs: d_in[0]=attention_mask f32[32*1*512*512], d_in[1]=bboxs f32[32*128*5],
//         d_in[2]=img_h i32[1], d_in[3]=img_w i32[1].  Output: f32[1].
// ---------------------------------------------------------------------------
extern "C" void kernel_launch(void* const* d_in, const int* in_sizes, int n_in,
                              void* d_out, int out_size, void* d_ws, size_t ws_size,
                              hipStream_t stream) {
  (void)in_sizes; (void)n_in; (void)out_size; (void)ws_size;
  const float* att   = (const float*)d_in[0];
  const float* bbox  = (const float*)d_in[1];
  const int*   imgh  = (const int*)d_in[2];
  const int*   imgw  = (const int*)d_in[3];
  float*       outp  = (float*)d_out;
  float*       parts = (float*)d_ws;     // B_IMG * N_TILES floats (4 KB), fully
                                         // rewritten every call -> deterministic

  dim3 grid1(N_TILES, B_IMG);
  attn_loss_main<<<grid1, 256, 0, stream>>>(att, bbox, imgh, imgw, parts);
  attn_loss_final<<<1, 1024, 0, stream>>>(bbox, imgh, imgw, parts, outp);
}